// NeuralODE_18382460027290
// MI455X (gfx1250) — compile-verified
//
#include <hip/hip_runtime.h>
#include <math.h>

// ---------------------------------------------------------------------------
// 4th-order Forest-Ruth / Yoshida symplectic integrator, elementwise over
// 4*1048576 independent (p,q) pendulum states:  q' = p,  p' = -sin(q).
//
// Compute-bound on v_sin_f32 (TRANS) + FP32 FMAs; WMMA/TDM do not apply
// (no matrix contraction, each element touched once). We target wave32
// VALU/TRANS co-execution: 4 independent lanes per thread keep 4 sins in
// flight and let the backend form v_pk_fma_f32 / VOPD pairs for the FMAs.
// ---------------------------------------------------------------------------

// Forest-Ruth coefficients (c3=c2, c4=c1, d3=d1, d4=0), exact double literals:
//   c13 = 2^(1/3) = 1.2599210498948731648 ; den = 2 - c13
#define FR_C1 0.675603595979828817    //  0.5 / den
#define FR_C2 (-0.175603595979828817) // (0.5 - 2^(-2/3)) / den
#define FR_D1 1.351207191959657634    //  1.0 / den
#define FR_D2 (-1.702414383919315268) // -c13 / den

#define EPS4 0.04f                    // 4 * eps, eps = 0.01

__global__ __launch_bounds__(256) void fr4_pendulum_kernel(
    const float* __restrict__ p0, const float* __restrict__ q0,
    const float* __restrict__ t0p, const float* __restrict__ t1p,
    float* __restrict__ out_p, float* __restrict__ out_q, int n)
{
    // t0/t1 are uniform -> scalar loads; n_steps computed on-device
    // (graph capture forbids host readback of device scalars).
    const float t0 = t0p[0];
    const float t1 = t1p[0];
    const float dt = t1 - t0;
    const int nsteps = (int)floorf(fabsf(dt) * (1.0f / EPS4) + 0.5f);
    const float h = (nsteps > 0) ? dt / (float)nsteps : 0.0f;

    // Hoist coefficient*h products out of the scan (uniform).
    const float c1h  = (float)FR_C1 * h;
    const float c2h  = (float)FR_C2 * h;
    const float nd1h = -(float)FR_D1 * h;   // negated: p = fma(nd, sin, p)
    const float nd2h = -(float)FR_D2 * h;

    const int tid  = blockIdx.x * blockDim.x + threadIdx.x;
    const int idx4 = tid * 4;

    if (idx4 + 3 < n) {
        // 128-bit vector loads (global_load_b128), 4 independent states.
        float4 pv = *(const float4*)(p0 + idx4);
        float4 qv = *(const float4*)(q0 + idx4);
        float pp[4] = {pv.x, pv.y, pv.z, pv.w};
        float qq[4] = {qv.x, qv.y, qv.z, qv.w};

        for (int s = 0; s < nsteps; ++s) {
            // substep 1: c1, d1
            #pragma unroll
            for (int l = 0; l < 4; ++l) qq[l] = fmaf(c1h, pp[l], qq[l]);
            #pragma unroll
            for (int l = 0; l < 4; ++l) pp[l] = fmaf(nd1h, __sinf(qq[l]), pp[l]);
            // substep 2: c2, d2
            #pragma unroll
            for (int l = 0; l < 4; ++l) qq[l] = fmaf(c2h, pp[l], qq[l]);
            #pragma unroll
            for (int l = 0; l < 4; ++l) pp[l] = fmaf(nd2h, __sinf(qq[l]), pp[l]);
            // substep 3: c3 = c2, d3 = d1
            #pragma unroll
            for (int l = 0; l < 4; ++l) qq[l] = fmaf(c2h, pp[l], qq[l]);
            #pragma unroll
            for (int l = 0; l < 4; ++l) pp[l] = fmaf(nd1h, __sinf(qq[l]), pp[l]);
            // substep 4: c4 = c1, d4 = 0  (no sin, no p update)
            #pragma unroll
            for (int l = 0; l < 4; ++l) qq[l] = fmaf(c1h, pp[l], qq[l]);
        }

        *(float4*)(out_p + idx4) = make_float4(pp[0], pp[1], pp[2], pp[3]);
        *(float4*)(out_q + idx4) = make_float4(qq[0], qq[1], qq[2], qq[3]);
    } else if (idx4 < n) {
        // Scalar tail (n not multiple of 4) — same math.
        for (int i = idx4; i < n; ++i) {
            float p = p0[i], q = q0[i];
            for (int s = 0; s < nsteps; ++s) {
                q = fmaf(c1h, p, q);
                p = fmaf(nd1h, __sinf(q), p);
                q = fmaf(c2h, p, q);
                p = fmaf(nd2h, __sinf(q), p);
                q = fmaf(c2h, p, q);
                p = fmaf(nd1h, __sinf(q), p);
                q = fmaf(c1h, p, q);
            }
            out_p[i] = p;
            out_q[i] = q;
        }
    }
}

extern "C" void kernel_launch(void* const* d_in, const int* in_sizes, int n_in,
                              void* d_out, int out_size, void* d_ws, size_t ws_size,
                              hipStream_t stream) {
    (void)n_in; (void)d_ws; (void)ws_size; (void)out_size;

    const float* p0 = (const float*)d_in[0];
    const float* q0 = (const float*)d_in[1];
    const float* t0 = (const float*)d_in[2];
    const float* t1 = (const float*)d_in[3];

    const int n = in_sizes[0];            // 4 * 1048576 elements per state
    float* out_p = (float*)d_out;         // tuple return: kp first,
    float* out_q = (float*)d_out + n;     // then kq

    const int threads = 256;
    const int work    = (n + 3) / 4;      // 4 elements per thread
    const int blocks  = (work + threads - 1) / threads;

    fr4_pendulum_kernel<<<blocks, threads, 0, stream>>>(
        p0, q0, t0, t1, out_p, out_q, n);
}